// MultiGranularityTokenCRFModel_37701222924702
// MI455X (gfx1250) — compile-verified
//
#include <hip/hip_runtime.h>
#include <math.h>

// Shapes from setup_inputs (fixed)
#define B_   64
#define S_   512
#define H_   768
#define C_   3
#define MS_  32
#define NTOK (B_ * S_)          // 32768 tokens
#define NSEG (B_ * MS_)         // 2048 segments
#define EMIT_ELEMS (NTOK * C_)  // 98304 emission floats

typedef float v2f __attribute__((ext_vector_type(2)));
typedef float v8f __attribute__((ext_vector_type(8)));

// ---------------------------------------------------------------------------
// Kernel A: one pass over hidden_states.
// Each wave (32 lanes) owns a 16-token tile and computes, via f32 WMMA,
// D[16x16] = A[16x768-chunked] x Wcomb[768x16] where
//   Wcomb col 0..2 = w_tok, col 3 = w_sent, cols 4..15 = 0.
// Cols 0..2 -> raw emissions (to d_out), col 3 -> per-token sentence
// projection t = h.w_sent, atomically segment-summed (mean pooling is linear,
// so pooling the scalar projection == projecting the pooled mean).
// ---------------------------------------------------------------------------
__global__ __launch_bounds__(256) void fused_matvec_pool(
    const float* __restrict__ hs,      // [NTOK, H]
    const int*   __restrict__ sids,    // [NTOK]
    const float* __restrict__ w_tok,   // [H, C]
    const float* __restrict__ w_sent,  // [H]
    float*       __restrict__ emis,    // [NTOK, C] raw (bias+gate applied later)
    float*       __restrict__ segsum,  // [NSEG]
    float*       __restrict__ segcnt)  // [NSEG]
{
    __shared__ float ldsW[H_ * 4];  // combined weights, layout [k][n], n in 0..3

    const int tid = threadIdx.x;
    // Stage combined weights (12 KB) once per block, coalesced.
    for (int idx = tid; idx < H_ * 4; idx += 256) {
        const int k = idx >> 2, n = idx & 3;
        ldsW[idx] = (n < 3) ? w_tok[k * C_ + n] : w_sent[k];
    }
    __syncthreads();

    const int lane   = tid & 31;
    const int wave   = tid >> 5;
    const int tile   = blockIdx.x * 8 + wave;   // 256 blocks * 8 waves = 2048 tiles
    const int tokBase = tile * 16;
    const int m    = lane & 15;        // A: row M, B: col N, per ISA layouts
    const int h2   = lane >> 4;        // half-wave selector
    const int koff = 2 * h2;           // K offset for this half (A/B VGPR striping)
    const int col  = m & 3;            // clamped Wcomb column (branch-free)
    const float msk = (m < 4) ? 1.0f : 0.0f;

    const float* rowPtr = hs + (size_t)(tokBase + m) * H_;
    __builtin_prefetch(rowPtr, 0, 0);            // global_prefetch_b8
    __builtin_prefetch(rowPtr + 384, 0, 0);

    v8f acc = {};
    // K loop: 192 x V_WMMA_F32_16X16X4_F32, accumulate on C (no D->A/B hazard).
    // All control flow uniform, no EXEC divergence: WMMA requires EXEC == all 1s.
#pragma unroll 4
    for (int k0 = 0; k0 < H_; k0 += 4) {
        // A fragment: lane holds (M=m, K=k0+koff .. k0+koff+1) -> aligned float2
        const float2 av = *(const float2*)(rowPtr + k0 + koff);
        v2f a; a.x = av.x; a.y = av.y;
        // B fragment: lane holds (N=m, K=k0+koff+j); zero for n>=4 via mask mult
        v2f b;
        b.x = ldsW[(k0 + koff    ) * 4 + col] * msk;
        b.y = ldsW[(k0 + koff + 1) * 4 + col] * msk;
        acc = __builtin_amdgcn_wmma_f32_16x16x4_f32(
            /*neg_a=*/false, a, /*neg_b=*/false, b,
            /*c_mod=*/(short)0, acc, /*reuse_a=*/false, /*reuse_b=*/false);
    }

    // D layout: lane l, VGPR r holds D[M = r + 8*h2][N = l%16]
    if (m < 3) {
        // token-classifier columns -> raw emissions
#pragma unroll
        for (int r = 0; r < 8; ++r) {
            const int token = tokBase + r + 8 * h2;
            emis[token * C_ + m] = acc[r];
        }
    } else if (m == 3) {
        // sentence projection column -> segment sums / counts
#pragma unroll
        for (int r = 0; r < 8; ++r) {
            const int token = tokBase + r + 8 * h2;
            const int id = sids[token];
            if (id >= 0) {
                const int seg = (token >> 9) * MS_ + id;  // token/S_ * MS_ + id
                atomicAdd(&segsum[seg], acc[r]);
                atomicAdd(&segcnt[seg], 1.0f);
            }
        }
    }
}

// ---------------------------------------------------------------------------
// Kernel B: per-segment sigmoid gate (2048 segments — trivial).
// ---------------------------------------------------------------------------
__global__ void seg_gate(const float* __restrict__ segsum,
                         const float* __restrict__ segcnt,
                         const float* __restrict__ b_sent,
                         float* __restrict__ gate_seg)
{
    const int i = blockIdx.x * blockDim.x + threadIdx.x;
    if (i < NSEG) {
        const float cnt  = segcnt[i];
        const float mean = segsum[i] / fmaxf(cnt, 1.0f);
        const float lg   = mean + b_sent[0];
        gate_seg[i] = 1.0f / (1.0f + __expf(-lg));
    }
}

// ---------------------------------------------------------------------------
// Kernel C: scatter gates to tokens, add b_tok, gate emissions in place.
// Emissions region of d_out is fully rewritten by kernel A every call, so the
// in-place multiply is graph-replay deterministic.
// ---------------------------------------------------------------------------
__global__ void apply_gates(const int* __restrict__ sids,
                            const float* __restrict__ gate_seg,
                            const float* __restrict__ b_tok,
                            float* __restrict__ emis,      // [NTOK, C]
                            float* __restrict__ gates_out) // [NTOK]
{
    const int t = blockIdx.x * blockDim.x + threadIdx.x;
    if (t < NTOK) {
        const int id = sids[t];
        const float g = (id >= 0) ? gate_seg[(t >> 9) * MS_ + id] : 1.0f;
        gates_out[t] = g;
#pragma unroll
        for (int c = 0; c < C_; ++c) {
            const int idx = t * C_ + c;
            emis[idx] = (emis[idx] + b_tok[c]) * g;
        }
    }
}

// ---------------------------------------------------------------------------
// Host-side launch (graph-capture safe: only async ops on `stream`).
// ws layout: [0,NSEG) segsum | [NSEG,2*NSEG) segcnt | [2*NSEG,3*NSEG) gate_seg
// ---------------------------------------------------------------------------
extern "C" void kernel_launch(void* const* d_in, const int* in_sizes, int n_in,
                              void* d_out, int out_size, void* d_ws, size_t ws_size,
                              hipStream_t stream) {
    const float* hs     = (const float*)d_in[0];
    const int*   sids   = (const int*)  d_in[1];
    const float* w_sent = (const float*)d_in[2];
    const float* b_sent = (const float*)d_in[3];
    const float* w_tok  = (const float*)d_in[4];
    const float* b_tok  = (const float*)d_in[5];

    float* emis  = (float*)d_out;            // [NTOK*C]
    float* gates = emis + EMIT_ELEMS;        // [NTOK]

    float* segsum   = (float*)d_ws;
    float* segcnt   = segsum + NSEG;
    float* gate_seg = segcnt + NSEG;

    // Zero accumulators every call (ws is poisoned once, never re-zeroed).
    hipMemsetAsync(d_ws, 0, 2 * NSEG * sizeof(float), stream);

    // 2048 tiles of 16 tokens, 8 waves/block -> 256 blocks.
    fused_matvec_pool<<<NTOK / (16 * 8), 256, 0, stream>>>(
        hs, sids, w_tok, w_sent, emis, segsum, segcnt);

    seg_gate<<<(NSEG + 255) / 256, 256, 0, stream>>>(segsum, segcnt, b_sent, gate_seg);

    apply_gates<<<(NTOK + 255) / 256, 256, 0, stream>>>(sids, gate_seg, b_tok, emis, gates);
}